// HyperAttention_31731218383034
// MI455X (gfx1250) — compile-verified
//
#include <hip/hip_runtime.h>
#include <math.h>

#define B_ 2
#define H_ 16
#define N_ 8192
#define D_ 64
#define NPROJ 7
#define BLK 256
#define NB (N_ / BLK)          /* 32 key blocks per head */
#define SAMPLE_ 256
#define BHN (B_ * H_ * N_)     /* 262144 rows */

typedef __attribute__((ext_vector_type(2))) float v2f;
typedef __attribute__((ext_vector_type(8))) float v8f;

static __device__ __forceinline__ v8f wmma_f32_16x16x4(v2f a, v2f b, v8f c) {
  // D(16x16,f32) += A(16x4,f32) * B(4x16,f32); exact fp32 path on CDNA5
  return __builtin_amdgcn_wmma_f32_16x16x4_f32(false, a, false, b, (short)0, c,
                                               false, false);
}

// ---------------------------------------------------------------------------
// K1: per-(b,h) max ||k||^2
// ---------------------------------------------------------------------------
__global__ void kmax_kernel(const float* __restrict__ key,
                            float* __restrict__ kmax2) {
  int bh = blockIdx.x;
  int tid = threadIdx.x;
  float mx = 0.f;
  for (int n = tid; n < N_; n += 256) {
    const float* kp = key + ((size_t)bh * N_ + n) * D_;
    float s = 0.f;
#pragma unroll
    for (int d = 0; d < D_; ++d) s += kp[d] * kp[d];
    mx = fmaxf(mx, s);
  }
  __shared__ float red[256];
  red[tid] = mx;
  __syncthreads();
  for (int off = 128; off > 0; off >>= 1) {
    if (tid < off) red[tid] = fmaxf(red[tid], red[tid + off]);
    __syncthreads();
  }
  if (tid == 0) kmax2[bh] = red[0];
}

// ---------------------------------------------------------------------------
// K2: angular LSH hashes (Gray-coded) for augmented queries and keys
// ---------------------------------------------------------------------------
__global__ void hash_kernel(const float* __restrict__ query,
                            const float* __restrict__ key,
                            const float* __restrict__ proj,
                            const float* __restrict__ kmax2,
                            int* __restrict__ qh, int* __restrict__ kh) {
  int tid = blockIdx.x * blockDim.x + threadIdx.x;
  if (tid >= BHN) return;
  int bh = tid / N_;
  float accq[NPROJ], acck[NPROJ];
#pragma unroll
  for (int r = 0; r < NPROJ; ++r) { accq[r] = 0.f; acck[r] = 0.f; }
  const float* qp = query + (size_t)tid * D_;
  const float* kp = key + (size_t)tid * D_;
  float n2 = 0.f;
  for (int d = 0; d < D_; ++d) {
    float qv = qp[d], kv = kp[d];
    n2 += kv * kv;
#pragma unroll
    for (int r = 0; r < NPROJ; ++r) {
      accq[r] += qv * proj[d * NPROJ + r];
      acck[r] += kv * proj[d * NPROJ + r];
    }
  }
  float extra = sqrtf(fmaxf(kmax2[bh] - n2, 0.f));
  int bq = 0, bk = 0;
#pragma unroll
  for (int r = 0; r < NPROJ; ++r) {
    acck[r] += extra * proj[D_ * NPROJ + r];
    bq |= (accq[r] > 0.f) ? (1 << r) : 0;
    bk |= (acck[r] > 0.f) ? (1 << r) : 0;
  }
  qh[tid] = bq ^ (bq >> 1);  // binary-reflected Gray code == PERM lookup
  kh[tid] = bk ^ (bk >> 1);
}

// ---------------------------------------------------------------------------
// K3: stable counting sort of 8192 7-bit hashes per (b,h). 64 chunks of 128
// elements each are counted/scattered serially per chunk => stability.
// ---------------------------------------------------------------------------
__global__ void lsh_sort_kernel(const int* __restrict__ qh,
                                const int* __restrict__ kh,
                                int* __restrict__ qidx,
                                int* __restrict__ kidx) {
  __shared__ int cnt[128 * 64];  // [bucket*64 + chunk], bucket-major
  __shared__ int partial[256];
  int which = blockIdx.x >> 5;
  int bh = blockIdx.x & 31;
  const int* h = (which ? kh : qh) + (size_t)bh * N_;
  int* out = (which ? kidx : qidx) + (size_t)bh * N_;
  int tid = threadIdx.x;

  for (int i = tid; i < 8192; i += 256) cnt[i] = 0;
  __syncthreads();
  if (tid < 64) {
    int base = tid * 128;
    for (int e = 0; e < 128; ++e) cnt[h[base + e] * 64 + tid]++;
  }
  __syncthreads();
  // exclusive scan over 8192 (bucket,chunk) entries
  int b0 = tid * 32;
  int s = 0;
  for (int i = 0; i < 32; ++i) s += cnt[b0 + i];
  partial[tid] = s;
  __syncthreads();
  if (tid == 0) {
    int run = 0;
    for (int i = 0; i < 256; ++i) { int t = partial[i]; partial[i] = run; run += t; }
  }
  __syncthreads();
  int run = partial[tid];
  for (int i = 0; i < 32; ++i) { int t = cnt[b0 + i]; cnt[b0 + i] = run; run += t; }
  __syncthreads();
  if (tid < 64) {
    int base = tid * 128;
    for (int e = 0; e < 128; ++e) {
      int b = h[base + e];
      int pos = cnt[b * 64 + tid]++;
      out[pos] = base + e;
    }
  }
}

// ---------------------------------------------------------------------------
// K4: block-diagonal attention over LSH-sorted blocks.
// One workgroup per (b,h,block); 16 waves; each wave owns a 16-query strip.
// Computes S^T = K*Q^T with V_WMMA_F32_16X16X4_F32, softmax, then P*V.
// ---------------------------------------------------------------------------
__global__ __launch_bounds__(512) void blockattn_kernel(
    const float* __restrict__ query, const float* __restrict__ key,
    const float* __restrict__ value, const int* __restrict__ qidx,
    const int* __restrict__ kidx, float* __restrict__ attn_blk,
    float* __restrict__ lse_blk) {
  int bh = blockIdx.x / NB;
  int g = blockIdx.x % NB;
  int tid = threadIdx.x;
  int wave = tid >> 5, lane = tid & 31, lo = lane & 15, hi = lane >> 4;
  const size_t base = (size_t)bh * N_;

  __shared__ float Vlds[BLK * D_];  // 64KB of the WGP's 320KB LDS
  for (int i = tid; i < BLK * D_; i += 512) {
    int r = i >> 6, d = i & 63;
    int row = kidx[base + g * BLK + r];
    Vlds[i] = value[(base + row) * D_ + d];
  }
  __syncthreads();

  // preload Q strip as the B operand of K*Q^T (float2 per lane per d-step)
  int qrow = qidx[base + g * BLK + wave * 16 + lo];
  const float* qp = query + (base + qrow) * D_ + hi * 2;
  v2f qv[16];
#pragma unroll
  for (int s = 0; s < 16; ++s) qv[s] = *(const v2f*)(qp + 4 * s);

  v8f st[16];  // S^T: 16 key-tiles x (16 keys x 16 queries)
#pragma unroll
  for (int t = 0; t < 16; ++t) {
    int krow = kidx[base + g * BLK + t * 16 + lo];
    const float* kp = key + (base + krow) * D_ + hi * 2;
    v8f acc = {0.f, 0.f, 0.f, 0.f, 0.f, 0.f, 0.f, 0.f};
#pragma unroll
    for (int s = 0; s < 16; ++s) {
      v2f a = *(const v2f*)(kp + 4 * s);
      acc = wmma_f32_16x16x4(a, qv[s], acc);
    }
    st[t] = acc;
  }

  const float scale = 0.125f;  // 1/sqrt(64)
  float m = -3.4e38f;
#pragma unroll
  for (int t = 0; t < 16; ++t)
#pragma unroll
    for (int v = 0; v < 8; ++v) {
      float x = st[t][v] * scale;
      st[t][v] = x;
      m = fmaxf(m, x);
    }
  m = fmaxf(m, __shfl_xor(m, 16, 32));
  float sum = 0.f;
#pragma unroll
  for (int t = 0; t < 16; ++t)
#pragma unroll
    for (int v = 0; v < 8; ++v) {
      float p = __expf(st[t][v] - m);
      st[t][v] = p;
      sum += p;
    }
  sum += __shfl_xor(sum, 16, 32);
  float lse = m + __logf(sum);
  if (lane < 16) lse_blk[base + g * BLK + wave * 16 + lane] = lse;

  // P*V: relayout S^T (C layout) into A chunks via half-wave broadcasts
  const v8f vz = {0.f, 0.f, 0.f, 0.f, 0.f, 0.f, 0.f, 0.f};
  v8f o[4] = {vz, vz, vz, vz};
#pragma unroll
  for (int cc = 0; cc < 64; ++cc) {
    int t = cc >> 2;
    int r = (cc & 3) * 4;
    int hs = (r >= 8) ? 1 : 0;
    float a0 = st[t][r & 7], a1 = st[t][(r & 7) + 1];
    float b0 = st[t][(r + 2) & 7], b1 = st[t][((r + 2) & 7) + 1];
    float a0o = __shfl_xor(a0, 16, 32), a1o = __shfl_xor(a1, 16, 32);
    float b0o = __shfl_xor(b0, 16, 32), b1o = __shfl_xor(b1, 16, 32);
    float a0s = (hi == hs) ? a0 : a0o, a1s = (hi == hs) ? a1 : a1o;
    float b0s = (hi == hs) ? b0 : b0o, b1s = (hi == hs) ? b1 : b1o;
    v2f A;
    A.x = hi ? b0s : a0s;
    A.y = hi ? b1s : a1s;
    int kc = cc * 4;
#pragma unroll
    for (int dt = 0; dt < 4; ++dt) {
      v2f Bv;
      Bv.x = Vlds[(kc + 2 * hi) * D_ + dt * 16 + lo];
      Bv.y = Vlds[(kc + 1 + 2 * hi) * D_ + dt * 16 + lo];
      o[dt] = wmma_f32_16x16x4(A, Bv, o[dt]);
    }
  }
  float inv[8];
#pragma unroll
  for (int v = 0; v < 8; ++v) inv[v] = 1.f / __shfl(sum, v + 8 * hi, 32);
#pragma unroll
  for (int dt = 0; dt < 4; ++dt)
#pragma unroll
    for (int v = 0; v < 8; ++v)
      attn_blk[(base + g * BLK + wave * 16 + v + 8 * hi) * D_ + dt * 16 + lo] =
          o[dt][v] * inv[v];
}

// ---------------------------------------------------------------------------
// K5: sampled-column residual attention + log-space merge + scatter unsort
// ---------------------------------------------------------------------------
__global__ __launch_bounds__(512) void residual_kernel(
    const float* __restrict__ query, const float* __restrict__ key,
    const float* __restrict__ value, const int* __restrict__ qidx,
    const int* __restrict__ kidx, const int* __restrict__ sampled,
    const float* __restrict__ attn_blk, const float* __restrict__ lse_blk,
    float* __restrict__ out) {
  int bh = blockIdx.x / NB;
  int g = blockIdx.x % NB;
  int tid = threadIdx.x;
  int wave = tid >> 5, lane = tid & 31, lo = lane & 15, hi = lane >> 4;
  const size_t base = (size_t)bh * N_;
  const int* smp = sampled + bh * SAMPLE_;

  __shared__ float Vlds[SAMPLE_ * D_];
  for (int i = tid; i < SAMPLE_ * D_; i += 512) {
    int r = i >> 6, d = i & 63;
    int row = kidx[base + smp[r]];
    Vlds[i] = value[(base + row) * D_ + d];
  }
  __syncthreads();

  int qrow = qidx[base + g * BLK + wave * 16 + lo];
  const float* qp = query + (base + qrow) * D_ + hi * 2;
  v2f qv[16];
#pragma unroll
  for (int s = 0; s < 16; ++s) qv[s] = *(const v2f*)(qp + 4 * s);

  v8f st[16];
#pragma unroll
  for (int t = 0; t < 16; ++t) {
    int krow = kidx[base + smp[t * 16 + lo]];
    const float* kp = key + (base + krow) * D_ + hi * 2;
    v8f acc = {0.f, 0.f, 0.f, 0.f, 0.f, 0.f, 0.f, 0.f};
#pragma unroll
    for (int s = 0; s < 16; ++s) {
      v2f a = *(const v2f*)(kp + 4 * s);
      acc = wmma_f32_16x16x4(a, qv[s], acc);
    }
    st[t] = acc;
  }

  const float scale = 0.125f;
  float m = -3.4e38f;
#pragma unroll
  for (int t = 0; t < 16; ++t)
#pragma unroll
    for (int v = 0; v < 8; ++v) {
      int kk = t * 16 + v + 8 * hi;  // sampled-column id this lane holds
      float bias = ((smp[kk] >> 8) == g) ? -3.4e38f : 0.f;
      float x = st[t][v] * scale + bias;
      st[t][v] = x;
      m = fmaxf(m, x);
    }
  m = fmaxf(m, __shfl_xor(m, 16, 32));
  float sum = 0.f;
#pragma unroll
  for (int t = 0; t < 16; ++t)
#pragma unroll
    for (int v = 0; v < 8; ++v) {
      float p = __expf(st[t][v] - m);
      st[t][v] = p;
      sum += p;
    }
  sum += __shfl_xor(sum, 16, 32);
  float lse_res = m + __logf(sum) + 3.4657359028f;  // + log(N/SAMPLE)

  const v8f vz = {0.f, 0.f, 0.f, 0.f, 0.f, 0.f, 0.f, 0.f};
  v8f o[4] = {vz, vz, vz, vz};
#pragma unroll
  for (int cc = 0; cc < 64; ++cc) {
    int t = cc >> 2;
    int r = (cc & 3) * 4;
    int hs = (r >= 8) ? 1 : 0;
    float a0 = st[t][r & 7], a1 = st[t][(r & 7) + 1];
    float b0 = st[t][(r + 2) & 7], b1 = st[t][((r + 2) & 7) + 1];
    float a0o = __shfl_xor(a0, 16, 32), a1o = __shfl_xor(a1, 16, 32);
    float b0o = __shfl_xor(b0, 16, 32), b1o = __shfl_xor(b1, 16, 32);
    float a0s = (hi == hs) ? a0 : a0o, a1s = (hi == hs) ? a1 : a1o;
    float b0s = (hi == hs) ? b0 : b0o, b1s = (hi == hs) ? b1 : b1o;
    v2f A;
    A.x = hi ? b0s : a0s;
    A.y = hi ? b1s : a1s;
    int kc = cc * 4;
#pragma unroll
    for (int dt = 0; dt < 4; ++dt) {
      v2f Bv;
      Bv.x = Vlds[(kc + 2 * hi) * D_ + dt * 16 + lo];
      Bv.y = Vlds[(kc + 1 + 2 * hi) * D_ + dt * 16 + lo];
      o[dt] = wmma_f32_16x16x4(A, Bv, o[dt]);
    }
  }

#pragma unroll
  for (int v = 0; v < 8; ++v) {
    int q = v + 8 * hi;
    size_t jrow = base + g * BLK + wave * 16 + q;
    float lseb = lse_blk[jrow];
    float lser = __shfl(lse_res, q, 32);
    float invs = 1.f / __shfl(sum, q, 32);
    float c = 1.f / (1.f + __expf(lser - lseb));
    int orow = qidx[jrow];
#pragma unroll
    for (int dt = 0; dt < 4; ++dt) {
      float ab = attn_blk[jrow * D_ + dt * 16 + lo];
      float ar = o[dt][v] * invs;
      out[(base + orow) * D_ + dt * 16 + lo] = c * ab + (1.f - c) * ar;
    }
  }
}

// ---------------------------------------------------------------------------
extern "C" void kernel_launch(void* const* d_in, const int* in_sizes, int n_in,
                              void* d_out, int out_size, void* d_ws,
                              size_t ws_size, hipStream_t stream) {
  const float* query = (const float*)d_in[0];
  const float* key = (const float*)d_in[1];
  const float* value = (const float*)d_in[2];
  const float* proj = (const float*)d_in[3];
  const int* sampled = (const int*)d_in[4];
  float* out = (float*)d_out;

  char* ws = (char*)d_ws;
  float* attn_blk = (float*)ws; ws += (size_t)BHN * D_ * sizeof(float);
  float* lse_blk = (float*)ws;  ws += (size_t)BHN * sizeof(float);
  int* qidx = (int*)ws;         ws += (size_t)BHN * sizeof(int);
  int* kidx = (int*)ws;         ws += (size_t)BHN * sizeof(int);
  int* qh = (int*)ws;           ws += (size_t)BHN * sizeof(int);
  int* kh = (int*)ws;           ws += (size_t)BHN * sizeof(int);
  float* kmax2 = (float*)ws;

  kmax_kernel<<<B_ * H_, 256, 0, stream>>>(key, kmax2);
  hash_kernel<<<BHN / 256, 256, 0, stream>>>(query, key, proj, kmax2, qh, kh);
  lsh_sort_kernel<<<2 * B_ * H_, 256, 0, stream>>>(qh, kh, qidx, kidx);
  blockattn_kernel<<<B_ * H_ * NB, 512, 0, stream>>>(query, key, value, qidx,
                                                     kidx, attn_blk, lse_blk);
  residual_kernel<<<B_ * H_ * NB, 512, 0, stream>>>(
      query, key, value, qidx, kidx, sampled, attn_blk, lse_blk, out);
}